// QuatCapsuleLayer_44023414784335
// MI455X (gfx1250) — compile-verified
//
#include <hip/hip_runtime.h>
#include <math.h>

typedef __attribute__((ext_vector_type(2))) float v2f;
typedef __attribute__((ext_vector_type(8))) float v8f;
typedef int v4i_vs __attribute__((vector_size(16)));   // matches async-LDS builtin proto

#define EPSV 1e-8f

#if defined(__has_builtin)
#if __has_builtin(__builtin_amdgcn_global_load_async_to_lds_b128) && \
    __has_builtin(__builtin_amdgcn_s_wait_asynccnt)
#define USE_ASYNC_LDS 1
#endif
#endif

// ---------------- zero fill (acc + deg) ----------------
__global__ void zero_ws_kernel(float* __restrict__ p, int n) {
    int i = blockIdx.x * blockDim.x + threadIdx.x;
    int stride = gridDim.x * blockDim.x;
    for (; i < n; i += stride) p[i] = 0.0f;
}

// ---------------- normalize the 32*8 learned quaternions ----------------
__global__ void qnorm_kernel(const float* __restrict__ q, float* __restrict__ qn) {
    int t = threadIdx.x;                 // 0..255 == O*I
    float w = q[t * 4 + 0], x = q[t * 4 + 1], y = q[t * 4 + 2], z = q[t * 4 + 3];
    float inv = 1.0f / (sqrtf(w * w + x * x + y * y + z * z) + EPSV);
    qn[t * 4 + 0] = w * inv;
    qn[t * 4 + 1] = x * inv;
    qn[t * 4 + 2] = y * inv;
    qn[t * 4 + 3] = z * inv;
}

// ---------------- spline-conv edge messages, weights staged in LDS ----------------
// weight: [O=32][W=125][C=8] = 32000 floats = 128000 bytes  (fits in 320KB WGP LDS)
__global__ __launch_bounds__(256) void edge_kernel(
    const float* __restrict__ x,          // [N,8]
    const float* __restrict__ pseudo,     // [E,3]
    const float* __restrict__ weight,     // [32,125,8]
    const int*   __restrict__ edge_index, // [2,E]
    float* __restrict__ acc,              // [N,32]
    float* __restrict__ deg,              // [N]
    int E) {
    __shared__ float wl[32 * 125 * 8];

#ifdef USE_ASYNC_LDS
    // CDNA5 async global->LDS engine: 8000 x b128, tracked by ASYNCcnt.
    for (int j = threadIdx.x; j < 8000; j += 256) {
        __builtin_amdgcn_global_load_async_to_lds_b128(
            (__attribute__((address_space(1))) v4i_vs*)(weight + (size_t)j * 4),
            (__attribute__((address_space(3))) v4i_vs*)(wl + (size_t)j * 4),
            0, 0);
    }
    __builtin_amdgcn_s_wait_asynccnt(0);
#else
    const float4* wg4 = (const float4*)weight;
    float4* wl4 = (float4*)wl;
    for (int j = threadIdx.x; j < 8000; j += 256) wl4[j] = wg4[j];
#endif
    __syncthreads();

    int e = blockIdx.x * blockDim.x + threadIdx.x;
    if (e >= E) return;

    // degree-1 open B-spline basis (K=5 -> scale by 4, i0 in [0,3])
    float sc0 = pseudo[e * 3 + 0] * 4.0f;
    float sc1 = pseudo[e * 3 + 1] * 4.0f;
    float sc2 = pseudo[e * 3 + 2] * 4.0f;
    float f0 = fminf(fmaxf(floorf(sc0), 0.0f), 3.0f);
    float f1 = fminf(fmaxf(floorf(sc1), 0.0f), 3.0f);
    float f2 = fminf(fmaxf(floorf(sc2), 0.0f), 3.0f);
    float fr0 = sc0 - f0, fr1 = sc1 - f1, fr2 = sc2 - f2;
    int i0 = (int)f0, i1 = (int)f1, i2 = (int)f2;

    int src = edge_index[e];
    const float4* xp = (const float4*)(x + (size_t)src * 8);
    float4 xa = xp[0];
    float4 xb = xp[1];

    float msg[32];
#pragma unroll
    for (int o = 0; o < 32; ++o) msg[o] = 0.0f;

    const float4* wp4 = (const float4*)wl;   // [o*250 + w*2 (+1)]

#pragma unroll
    for (int bcorner = 0; bcorner < 8; ++bcorner) {
        int b0 = bcorner & 1, b1 = (bcorner >> 1) & 1, b2 = (bcorner >> 2) & 1;
        int w = (i0 + b0) + 5 * (i1 + b1) + 25 * (i2 + b2);
        float bas = (b0 ? fr0 : 1.0f - fr0) *
                    (b1 ? fr1 : 1.0f - fr1) *
                    (b2 ? fr2 : 1.0f - fr2);
        int wbase = w * 2;
#pragma unroll 4
        for (int o = 0; o < 32; ++o) {
            float4 wa = wp4[o * 250 + wbase];
            float4 wb = wp4[o * 250 + wbase + 1];
            float t = wa.x * xa.x + wa.y * xa.y + wa.z * xa.z + wa.w * xa.w +
                      wb.x * xb.x + wb.y * xb.y + wb.z * xb.z + wb.w * xb.w;
            msg[o] += bas * t;
        }
    }

    int dst = edge_index[E + e];
    float* accp = acc + (size_t)dst * 32;
#pragma unroll
    for (int o = 0; o < 32; ++o) atomicAdd(accp + o, msg[o]);
    atomicAdd(deg + dst, 1.0f);
}

// ---------------- root term x @ root^T via V_WMMA_F32_16X16X4_F32 ----------------
// A: 16 nodes x K (x), B: K x 16 outputs (root^T). K=8 -> two K=4 WMMA steps.
// Two o-tiles (O=32) per wave. N % 16 == 0 for this problem (50000/16 = 3125).
__global__ __launch_bounds__(256) void root_wmma_kernel(
    const float* __restrict__ x,     // [N,8]
    const float* __restrict__ root,  // [32,8]
    float* __restrict__ rootterm,    // [N,32]
    int N) {
    int wave = (blockIdx.x * blockDim.x + threadIdx.x) >> 5;
    int lane = threadIdx.x & 31;
    int tiles = (N + 15) >> 4;
    if (wave >= tiles) return;                 // wave-uniform: EXEC all-ones below

    int node_base = wave << 4;
    int m = lane & 15;                          // row-within-tile / column index
    int h = lane >> 4;                          // lane half selects K pair
    int nodeA = node_base + m;
    if (nodeA > N - 1) nodeA = N - 1;           // clamp (branchless)

    v8f c0 = {}; // o = 0..15
    v8f c1 = {}; // o = 16..31
#pragma unroll
    for (int ks = 0; ks < 2; ++ks) {
        int kbase = 2 * h + 4 * ks;             // ISA layout: K = r + 2*(lane/16)
        v2f a, b0, b1;
        a[0]  = x[(size_t)nodeA * 8 + kbase + 0];
        a[1]  = x[(size_t)nodeA * 8 + kbase + 1];
        b0[0] = root[(size_t)m * 8 + kbase + 0];        // B[k][o] = root[o][k]
        b0[1] = root[(size_t)m * 8 + kbase + 1];
        b1[0] = root[(size_t)(16 + m) * 8 + kbase + 0];
        b1[1] = root[(size_t)(16 + m) * 8 + kbase + 1];
        c0 = __builtin_amdgcn_wmma_f32_16x16x4_f32(false, a, false, b0,
                                                   (short)0, c0, false, false);
        c1 = __builtin_amdgcn_wmma_f32_16x16x4_f32(false, a, false, b1,
                                                   (short)0, c1, false, false);
    }

    // C: VGPR j holds row M = j + 8*h, column m. Full-tile test is wave-uniform,
    // so the common path is an unguarded store clause (no exec-mask churn).
    if (node_base + 16 <= N) {
#pragma unroll
        for (int j = 0; j < 8; ++j) {
            int nr = node_base + j + 8 * h;
            rootterm[(size_t)nr * 32 + m]      = c0[j];
            rootterm[(size_t)nr * 32 + 16 + m] = c1[j];
        }
    } else {
#pragma unroll
        for (int j = 0; j < 8; ++j) {
            int nr = node_base + j + 8 * h;
            if (nr < N) {
                rootterm[(size_t)nr * 32 + m]      = c0[j];
                rootterm[(size_t)nr * 32 + 16 + m] = c1[j];
            }
        }
    }
}

// ---------------- per-node: conv_act + quaternion votes + routing + output ----------
// One wave32 per node, lane = output capsule o. Softmax over O == butterfly shfl_xor.
__global__ __launch_bounds__(256) void node_kernel(
    const float* __restrict__ pose,      // [N,8,4] (unit quaternions)
    const float* __restrict__ acc,       // [N,32]
    const float* __restrict__ degp,      // [N]
    const float* __restrict__ rootterm,  // [N,32]
    const float* __restrict__ qn,        // [32,8,4] normalized
    const float* __restrict__ alpha,     // [32]
    const float* __restrict__ beta,      // [32]
    const float* __restrict__ bias,      // [32]
    float* __restrict__ out,             // [N,32,5]
    int N) {
    int n = (blockIdx.x * blockDim.x + threadIdx.x) >> 5;
    int o = threadIdx.x & 31;
    if (n >= N) return;

    float d = fmaxf(degp[n], 1.0f);
    float conv = acc[(size_t)n * 32 + o] / d + rootterm[(size_t)n * 32 + o] + bias[o];

    // votes v[o][i] = qmul(qn[o][i], pose[n][i])
    float v[8][4];
    const float4* qp = (const float4*)qn;
    const float4* pp = (const float4*)pose;
#pragma unroll
    for (int i = 0; i < 8; ++i) {
        float4 q = qp[o * 8 + i];
        float4 p = pp[(size_t)n * 8 + i];
        v[i][0] = q.x * p.x - q.y * p.y - q.z * p.z - q.w * p.w;
        v[i][1] = q.x * p.y + q.y * p.x + q.z * p.w - q.w * p.z;
        v[i][2] = q.x * p.z - q.y * p.w + q.z * p.x + q.w * p.y;
        v[i][3] = q.x * p.w + q.y * p.z - q.z * p.y + q.w * p.x;
    }

    float blog[8];
#pragma unroll
    for (int i = 0; i < 8; ++i) blog[i] = 0.0f;
    float po0 = 0.f, po1 = 0.f, po2 = 0.f, po3 = 0.f;

    for (int it = 0; it < 3; ++it) {
        float s0 = 0.f, s1 = 0.f, s2 = 0.f, s3 = 0.f;
#pragma unroll
        for (int i = 0; i < 8; ++i) {
            float eb = __expf(blog[i]);
            float sum = eb;
#pragma unroll
            for (int off = 16; off >= 1; off >>= 1)
                sum += __shfl_xor(sum, off, 32);     // softmax denom over all 32 o-lanes
            float ci = eb / sum;
            s0 += ci * v[i][0];
            s1 += ci * v[i][1];
            s2 += ci * v[i][2];
            s3 += ci * v[i][3];
        }
        float inv = 1.0f / (sqrtf(s0 * s0 + s1 * s1 + s2 * s2 + s3 * s3) + EPSV);
        po0 = s0 * inv; po1 = s1 * inv; po2 = s2 * inv; po3 = s3 * inv;
#pragma unroll
        for (int i = 0; i < 8; ++i)
            blog[i] += v[i][0] * po0 + v[i][1] * po1 + v[i][2] * po2 + v[i][3] * po3;
    }

    float act = 1.0f / (1.0f + __expf(-(alpha[o] * conv + beta[o])));
    float* op = out + ((size_t)n * 32 + o) * 5;
    op[0] = po0; op[1] = po1; op[2] = po2; op[3] = po3; op[4] = act;
}

extern "C" void kernel_launch(void* const* d_in, const int* in_sizes, int n_in,
                              void* d_out, int out_size, void* d_ws, size_t ws_size,
                              hipStream_t stream) {
    (void)n_in; (void)out_size; (void)ws_size;
    const float* pose        = (const float*)d_in[0];
    const float* x           = (const float*)d_in[1];
    const float* pseudo      = (const float*)d_in[2];
    const float* quaternions = (const float*)d_in[3];
    const float* alpha       = (const float*)d_in[4];
    const float* beta        = (const float*)d_in[5];
    const float* weight      = (const float*)d_in[6];
    const float* root        = (const float*)d_in[7];
    const float* bias        = (const float*)d_in[8];
    const int*   edge_index  = (const int*)d_in[9];

    int N = in_sizes[1] / 8;      // x: [N,8]
    int E = in_sizes[2] / 3;      // pseudo: [E,3]

    float* acc      = (float*)d_ws;                 // [N,32]
    float* deg      = acc + (size_t)N * 32;         // [N]
    float* rootterm = deg + N;                      // [N,32]
    float* qn       = rootterm + (size_t)N * 32;    // [32,8,4]

    int zcount = N * 33;                            // acc + deg contiguous
    zero_ws_kernel<<<(zcount + 255) / 256, 256, 0, stream>>>(acc, zcount);
    qnorm_kernel<<<1, 256, 0, stream>>>(quaternions, qn);
    edge_kernel<<<(E + 255) / 256, 256, 0, stream>>>(x, pseudo, weight, edge_index,
                                                     acc, deg, E);
    int tiles = (N + 15) / 16;
    root_wmma_kernel<<<(tiles + 7) / 8, 256, 0, stream>>>(x, root, rootterm, N);
    node_kernel<<<(N + 7) / 8, 256, 0, stream>>>(pose, acc, deg, rootterm, qn,
                                                 alpha, beta, bias, (float*)d_out, N);
}